// Grouper4_2903397892782
// MI455X (gfx1250) — compile-verified
//
#include <hip/hip_runtime.h>
#include <hip/hip_bf16.h>
#include <math.h>

// Problem constants (match reference)
#define B_ 2
#define N_ 16384
#define M_ 4096
#define C_ 16
#define K_ 32
#define RADIUS_ 0.1f
#define EPS_ 1e-8f

typedef __attribute__((ext_vector_type(2))) float v2f;
typedef __attribute__((ext_vector_type(8))) float v8f;

// ---------------------------------------------------------------------------
// Kernel 0: copy new_xyz (B,M,3) to the head of d_out
// ---------------------------------------------------------------------------
__global__ __launch_bounds__(256) void copy_newxyz_kernel(const float* __restrict__ nx,
                                                          float* __restrict__ out) {
    int i = blockIdx.x * 256 + threadIdx.x;
    if (i < B_ * M_ * 3) out[i] = nx[i];
}

// ---------------------------------------------------------------------------
// Kernel 1: ball query. One wave32 per query row. Ordered first-K selection
// via ballot + prefix popcount over consecutive 32-point chunks.
// ---------------------------------------------------------------------------
__global__ __launch_bounds__(256) void ballquery_kernel(const float* __restrict__ xyz,
                                                        const float* __restrict__ new_xyz,
                                                        int* __restrict__ idx_out,
                                                        int* __restrict__ cnt_out) {
    __shared__ int arr_s[8][K_];
    const int wave = threadIdx.x >> 5;
    const int lane = threadIdx.x & 31;
    volatile int* arr = arr_s[wave];          // wave-private; DS ops in-order per wave

    const int row = blockIdx.x * 8 + wave;    // row = b*M + m
    const int b = row / M_;
    const float qx = new_xyz[row * 3 + 0];
    const float qy = new_xyz[row * 3 + 1];
    const float qz = new_xyz[row * 3 + 2];
    const float* __restrict__ xb = xyz + (size_t)b * N_ * 3;
    const float r2 = RADIUS_ * RADIUS_;

    int cnt = 0;
    for (int base = 0; base < N_; base += 32) {
        const int n = base + lane;
        // stream xyz ahead (global_prefetch_b8, speculative RT)
        int pf = n + 256; pf = pf < N_ ? pf : (N_ - 1);
        __builtin_prefetch(xb + (size_t)pf * 3, 0, 1);

        const float dx = xb[n * 3 + 0] - qx;
        const float dy = xb[n * 3 + 1] - qy;
        const float dz = xb[n * 3 + 2] - qz;
        const bool hit = (dx * dx + dy * dy + dz * dz) < r2;
        const unsigned mask = (unsigned)__ballot(hit);   // wave32: low 32 bits
        const int rank = __popc(mask & ((1u << lane) - 1u));
        if (hit && (cnt + rank) < K_) arr[cnt + rank] = n;
        cnt += __popc(mask);
        if (cnt >= K_) { cnt = K_; break; }
    }
    // pad empty slots with first found index; empty row -> idx 0 (safe gather)
    const int v = (cnt == 0) ? 0 : ((lane < cnt) ? arr[lane] : arr[0]);
    idx_out[row * K_ + lane] = v;
    if (lane == 0) cnt_out[row] = cnt;
}

// ---------------------------------------------------------------------------
// Kernel 2: gather + SharedMLP + pooling. One wave32 per query row; 4 waves
// per block. All matmuls via V_WMMA_F32_16X16X4_F32 (f32 exact path).
//
// WMMA operand layouts (wave32, per ISA 7.12.2):
//   A 16x4 : lane l (l<16): M=l, regs {K=0,K=1}; lane l>=16: M=l-16, {K=2,K=3}
//   B 4x16 : lane l (l<16): N=l, regs {K=0,K=1}; lane l>=16: N=l-16, {K=2,K=3}
//   C/D    : reg g, lane l: (M = g + 8*(l/16), N = l%16)
// ---------------------------------------------------------------------------
#define WAVES2 4
__global__ __launch_bounds__(128) void group_mlp_kernel(
    const float* __restrict__ xyz, const float* __restrict__ new_xyz,
    const float* __restrict__ features,
    const float* __restrict__ w0, const float* __restrict__ b0,
    const float* __restrict__ w1, const float* __restrict__ b1,
    const float* __restrict__ wxyz, const float* __restrict__ bxyz,
    const int* __restrict__ idx_in, const int* __restrict__ cnt_in,
    float* __restrict__ out_feat /* = d_out + B*M*3, layout (B,96,M) */) {

    __shared__ float s_w0[32 * 16];
    __shared__ float s_w1[64 * 32];
    __shared__ float s_wx[32 * 4];        // wxyz padded to K=4 (col 3 = 0)
    __shared__ float s_b0[32], s_b1[64], s_bx[32];
    __shared__ float s_feat[WAVES2][16][32];   // [c][k]
    __shared__ float s_h0[WAVES2][32][32];     // [c][k]
    __shared__ float s_rel[WAVES2][4][32];     // [c][k], row 3 = 0

    const int t = threadIdx.x;

    // Block-cooperative weight load into LDS
    for (int i = t; i < 32 * 16; i += 128) s_w0[i] = w0[i];
    for (int i = t; i < 64 * 32; i += 128) s_w1[i] = w1[i];
    for (int i = t; i < 32 * 4; i += 128) {
        const int r = i >> 2, c = i & 3;
        s_wx[i] = (c < 3) ? wxyz[r * 3 + c] : 0.0f;
    }
    if (t < 32) s_b0[t] = b0[t];
    if (t < 64) s_b1[t] = b1[t];
    if (t < 32) s_bx[t] = bxyz[t];
    __syncthreads();

    const int wave = t >> 5, lane = t & 31;
    const int hi = lane >> 4;        // half-wave select
    const int lo = lane & 15;
    const int row = blockIdx.x * WAVES2 + wave;   // b*M + m
    const int b = row / M_, m = row % M_;

    // ---- Stage A: per-neighbor rel coords, dist-reciprocal weights --------
    const int myidx = idx_in[row * K_ + lane];    // lane == neighbor slot k
    const int cnt = cnt_in[row];
    const float qx = new_xyz[row * 3 + 0];
    const float qy = new_xyz[row * 3 + 1];
    const float qz = new_xyz[row * 3 + 2];
    const float* __restrict__ p = xyz + ((size_t)b * N_ + myidx) * 3;
    const float rx = p[0] - qx, ry = p[1] - qy, rz = p[2] - qz;
    s_rel[wave][0][lane] = rx;
    s_rel[wave][1][lane] = ry;
    s_rel[wave][2][lane] = rz;
    s_rel[wave][3][lane] = 0.0f;

    const float dist = sqrtf(rx * rx + ry * ry + rz * rz);
    // padding multiplicity: slot 0 and slots >= cnt are the same point
    const float idn = (cnt < K_ && (lane == 0 || lane >= cnt)) ? (float)(K_ - cnt + 1) : 1.0f;
    const float recip = (1.0f / (dist + EPS_)) / idn;
    float ssum = recip;
    #pragma unroll
    for (int off = 16; off > 0; off >>= 1) ssum += __shfl_xor(ssum, off, 32);
    const float weight = (cnt == 0) ? 0.0f : (recip / ssum);

    // ---- Stage B: gather 16-channel feature column for neighbor `lane` ----
    const float* __restrict__ fb = features + (size_t)b * C_ * N_;
    #pragma unroll
    for (int c = 0; c < C_; ++c)
        s_feat[wave][c][lane] = fb[(size_t)c * N_ + myidx];
    __syncthreads();

    // ---- Layer 0: h0(32x32) = relu(w0(32x16) @ feat(16x32) + b0) ----------
    #pragma unroll
    for (int mt = 0; mt < 2; ++mt) {
        #pragma unroll
        for (int nt = 0; nt < 2; ++nt) {
            v8f acc;
            #pragma unroll
            for (int g = 0; g < 8; ++g) acc[g] = s_b0[mt * 16 + g + 8 * hi];
            #pragma unroll
            for (int kc = 0; kc < 4; ++kc) {
                const int k0 = kc * 4 + 2 * hi;
                v2f a, bb;
                a.x = s_w0[(mt * 16 + lo) * 16 + k0];
                a.y = s_w0[(mt * 16 + lo) * 16 + k0 + 1];
                bb.x = s_feat[wave][k0][nt * 16 + lo];
                bb.y = s_feat[wave][k0 + 1][nt * 16 + lo];
                acc = __builtin_amdgcn_wmma_f32_16x16x4_f32(false, a, false, bb,
                                                            (short)0, acc, false, false);
            }
            #pragma unroll
            for (int g = 0; g < 8; ++g)
                s_h0[wave][mt * 16 + g + 8 * hi][nt * 16 + lo] = fmaxf(acc[g], 0.0f);
        }
    }
    __syncthreads();

    // ---- Layer 1: h1(64x32) = relu(w1 @ h0 + b1); weighted sum over K ----
    #pragma unroll
    for (int mt = 0; mt < 4; ++mt) {
        float fs[8];
        #pragma unroll
        for (int g = 0; g < 8; ++g) fs[g] = 0.0f;
        #pragma unroll
        for (int nt = 0; nt < 2; ++nt) {
            v8f acc;
            #pragma unroll
            for (int g = 0; g < 8; ++g) acc[g] = s_b1[mt * 16 + g + 8 * hi];
            #pragma unroll
            for (int kc = 0; kc < 8; ++kc) {
                const int k0 = kc * 4 + 2 * hi;
                v2f a, bb;
                a.x = s_w1[(mt * 16 + lo) * 32 + k0];
                a.y = s_w1[(mt * 16 + lo) * 32 + k0 + 1];
                bb.x = s_h0[wave][k0][nt * 16 + lo];
                bb.y = s_h0[wave][k0 + 1][nt * 16 + lo];
                acc = __builtin_amdgcn_wmma_f32_16x16x4_f32(false, a, false, bb,
                                                            (short)0, acc, false, false);
            }
            // neighbor weight for this lane's column in the D tile
            const float wk = __shfl(weight, nt * 16 + lo, 32);
            #pragma unroll
            for (int g = 0; g < 8; ++g) {
                float v = fmaxf(acc[g], 0.0f) * wk;
                v += __shfl_xor(v, 1, 32);
                v += __shfl_xor(v, 2, 32);
                v += __shfl_xor(v, 4, 32);
                v += __shfl_xor(v, 8, 32);   // sum over 16 columns of the half
                fs[g] += v;
            }
        }
        if (lo == 0) {
            #pragma unroll
            for (int g = 0; g < 8; ++g) {
                const int ch = 32 + mt * 16 + g + 8 * hi;  // feat_sum -> channels 32..95
                out_feat[((size_t)b * 96 + ch) * M_ + m] = fs[g];
            }
        }
    }

    // ---- xyz branch: hx(32x32) = relu(wxyz(32x3) @ rel + bxyz); max over K
    #pragma unroll
    for (int mt = 0; mt < 2; ++mt) {
        float fm[8];
        #pragma unroll
        for (int g = 0; g < 8; ++g) fm[g] = -3.4e38f;
        #pragma unroll
        for (int nt = 0; nt < 2; ++nt) {
            v8f acc;
            #pragma unroll
            for (int g = 0; g < 8; ++g) acc[g] = s_bx[mt * 16 + g + 8 * hi];
            const int k0 = 2 * hi;
            v2f a, bb;
            a.x = s_wx[(mt * 16 + lo) * 4 + k0];
            a.y = s_wx[(mt * 16 + lo) * 4 + k0 + 1];
            bb.x = s_rel[wave][k0][nt * 16 + lo];
            bb.y = s_rel[wave][k0 + 1][nt * 16 + lo];
            acc = __builtin_amdgcn_wmma_f32_16x16x4_f32(false, a, false, bb,
                                                        (short)0, acc, false, false);
            #pragma unroll
            for (int g = 0; g < 8; ++g) {
                float v = fmaxf(acc[g], 0.0f);
                v = fmaxf(v, __shfl_xor(v, 1, 32));
                v = fmaxf(v, __shfl_xor(v, 2, 32));
                v = fmaxf(v, __shfl_xor(v, 4, 32));
                v = fmaxf(v, __shfl_xor(v, 8, 32));
                fm[g] = fmaxf(fm[g], v);
            }
        }
        if (lo == 0) {
            #pragma unroll
            for (int g = 0; g < 8; ++g) {
                const int ch = mt * 16 + g + 8 * hi;       // feat_max -> channels 0..31
                out_feat[((size_t)b * 96 + ch) * M_ + m] = fm[g];
            }
        }
    }
}

// ---------------------------------------------------------------------------
extern "C" void kernel_launch(void* const* d_in, const int* in_sizes, int n_in,
                              void* d_out, int out_size, void* d_ws, size_t ws_size,
                              hipStream_t stream) {
    const float* xyz      = (const float*)d_in[0];
    const float* new_xyz  = (const float*)d_in[1];
    const float* features = (const float*)d_in[2];
    const float* w0   = (const float*)d_in[3];
    const float* b0   = (const float*)d_in[4];
    const float* w1   = (const float*)d_in[5];
    const float* b1   = (const float*)d_in[6];
    const float* wxyz = (const float*)d_in[7];
    const float* bxyz = (const float*)d_in[8];
    float* out = (float*)d_out;

    int* idx = (int*)d_ws;                 // B*M*K ints
    int* cnt = idx + (size_t)B_ * M_ * K_; // B*M ints

    copy_newxyz_kernel<<<(B_ * M_ * 3 + 255) / 256, 256, 0, stream>>>(new_xyz, out);
    ballquery_kernel<<<B_ * M_ / 8, 256, 0, stream>>>(xyz, new_xyz, idx, cnt);
    group_mlp_kernel<<<B_ * M_ / WAVES2, 128, 0, stream>>>(
        xyz, new_xyz, features, w0, b0, w1, b1, wxyz, bxyz,
        idx, cnt, out + (size_t)B_ * M_ * 3);
}